// ProximityAwareLoss4Class_25786983645937
// MI455X (gfx1250) — compile-verified
//
#include <hip/hip_runtime.h>
#include <hip/hip_bf16.h>
#include <stdint.h>

// ProximityAwareLoss4Class for MI455X (gfx1250).
// One workgroup (256 threads, 8 wave32) per batch row of S=8192 positions.
// Bandwidth-bound: logits read exactly once via coalesced b128 loads; labels
// streamed into LDS via CDNA5 async global->LDS copies (ASYNCcnt path).
// Sequential row dependencies (mode state machine, fwd/bwd nearest-distance)
// resolved with chunked scans: 32-position chunks per thread + 256-chunk
// cross-scan, distances saturated to 3 bits.

#define S_LEN 8192
#define NCH   256
#define CHUNK 32
#define TOLD  5

__device__ __forceinline__ int imin(int a, int b) { return a < b ? a : b; }

__global__ void pal_init_acc(float* acc) { acc[0] = 0.f; acc[1] = 0.f; }

__global__ void pal_finalize(const float* __restrict__ acc, float* __restrict__ out) {
    out[0] = acc[0] / fmaxf(acc[1], 1.0f);
}

__global__ __launch_bounds__(256)
void pal_row_kernel(const float* __restrict__ logits,
                    const int* __restrict__ labels,
                    float* __restrict__ g_acc)
{
    __shared__ int           s_lbl[S_LEN];        // 32 KB  (async-filled)
    __shared__ float         s_ce[S_LEN];         // 32 KB
    __shared__ unsigned char s_code[S_LEN];       //  8 KB  pred|valid|lab
    __shared__ unsigned char s_state[NCH];        // chunk summaries (mode SM)
    __shared__ unsigned char s_first[4][NCH];     // first True idx per mask (255=none)
    __shared__ unsigned char s_last[4][NCH];      // last True idx per mask  (255=none)
    __shared__ unsigned char s_inSt[NCH];         // incoming (pm,tm) per chunk
    __shared__ unsigned char s_fIn[4][NCH];       // incoming fwd dist per mask
    __shared__ unsigned char s_bIn[4][NCH];       // incoming bwd dist per mask
    __shared__ int           s_hasAny[4];
    __shared__ float         s_tab[8];            // predicted-switch multiplier by d_true
    __shared__ float         s_mode[4];           // mode-SM multiplier LUT
    __shared__ float         s_red[NCH];
    __shared__ float         s_redc[NCH];

    const int tid = threadIdx.x;
    const int row = blockIdx.x;

    if (tid == 0) {
        s_tab[0] = 0.1f;      // d_true == 0
        s_tab[1] = 0.7f;      // DECAY^1
        s_tab[2] = 0.49f;
        s_tab[3] = 0.343f;
        s_tab[4] = 0.2401f;
        s_tab[5] = 0.16807f;  // DECAY^5
        s_tab[6] = 10.0f;     // d_true > TOL -> FP_PEN
        s_tab[7] = 10.0f;
        s_mode[0] = 1.0f;     // neither
        s_mode[1] = 0.1f;     // good switch bonus
        s_mode[2] = 100.0f;   // invalid transition penalty
        s_mode[3] = 10.0f;    // both
    }

    // ---- Phase 0: CDNA5 async copy of this row's labels (32 KB) into LDS ----
    {
        const uint64_t gbase = (uint64_t)(uintptr_t)(labels + (size_t)row * S_LEN);
        const uint32_t lbase = (uint32_t)(uintptr_t)(&s_lbl[0]);  // low 32b of flat = LDS offset
#pragma unroll
        for (int i = 0; i < 8; ++i) {
            const int      byteoff = (i * 256 + tid) * 16;        // coalesced 16B per lane
            const uint32_t laddr   = lbase + (uint32_t)byteoff;
            const uint64_t gaddr   = gbase + (uint64_t)byteoff;
            asm volatile("global_load_async_to_lds_b128 %0, %1, off"
                         :: "v"(laddr), "v"(gaddr) : "memory");
        }
        asm volatile("s_wait_asynccnt 0" ::: "memory");
    }
    __syncthreads();

    // ---- Phase 1: stream logits once, fused argmax + weighted CE + code ----
#pragma unroll 4
    for (int i = 0; i < 32; ++i) {
        const int    pos = i * 256 + tid;                          // coalesced
        const float4 L   = ((const float4*)(logits + ((size_t)row * S_LEN + pos) * 4))[0];
        const int    lab = s_lbl[pos];
        const int    valid = (lab != -100) ? 1 : 0;
        const int    labs  = valid ? lab : 0;

        int   pred = 0;
        float best = L.x;
        if (L.y > best) { best = L.y; pred = 1; }
        if (L.z > best) { best = L.z; pred = 2; }
        if (L.w > best) { best = L.w; pred = 3; }

        const float e = __expf(L.x - best) + __expf(L.y - best) +
                        __expf(L.z - best) + __expf(L.w - best);
        const float lse = __logf(e) + best;
        const float lj  = (labs == 0) ? L.x : (labs == 1) ? L.y : (labs == 2) ? L.z : L.w;
        const float w   = (labs >= 2) ? 30.0f : 1.0f;

        s_ce[pos]   = (lse - lj) * w;
        s_code[pos] = (unsigned char)(pred | (valid << 2) | (labs << 3));
    }
    __syncthreads();

    // ---- Phase 2: per-chunk scan summaries ----
    {
        const int base = tid * CHUNK;
        int st = 0;  // bit0 hasValid, bit1 lastPredMode, bit2 lastLabMode
        int f0 = 255, f1 = 255, f2 = 255, f3 = 255;
        int l0 = 255, l1 = 255, l2 = 255, l3 = 255;
#pragma unroll
        for (int p = 0; p < CHUNK; ++p) {
            const int code  = s_code[base + p];
            const int pred  = code & 3;
            const int valid = (code >> 2) & 1;
            const int lab   = (code >> 3) & 3;
            if (valid) {
                const int pmode = (pred == 0 || pred == 2) ? 0 : 1;
                const int tmode = (lab == 0 || lab == 2) ? 0 : 1;
                st = 1 | (pmode << 1) | (tmode << 2);
            }
            if (pred == 2)            { if (f0 == 255) f0 = p; l0 = p; }
            if (pred == 3)            { if (f1 == 255) f1 = p; l1 = p; }
            if (valid && (lab == 2))  { if (f2 == 255) f2 = p; l2 = p; }
            if (valid && (lab == 3))  { if (f3 == 255) f3 = p; l3 = p; }
        }
        s_state[tid]    = (unsigned char)st;
        s_first[0][tid] = (unsigned char)f0;  s_last[0][tid] = (unsigned char)l0;
        s_first[1][tid] = (unsigned char)f1;  s_last[1][tid] = (unsigned char)l1;
        s_first[2][tid] = (unsigned char)f2;  s_last[2][tid] = (unsigned char)l2;
        s_first[3][tid] = (unsigned char)f3;  s_last[3][tid] = (unsigned char)l3;
    }
    __syncthreads();

    // ---- Phase 3: cross-chunk scans (9 threads, 256 cheap steps each) ----
    if (tid == 0) {
        int pm = 0, tm = 0;
        for (int c = 0; c < NCH; ++c) {
            s_inSt[c] = (unsigned char)(pm | (tm << 1));
            const int st = s_state[c];
            if (st & 1) { pm = (st >> 1) & 1; tm = (st >> 2) & 1; }
        }
    } else if (tid <= 4) {
        const int m = tid - 1;
        int f = 7, any = 0;
        for (int c = 0; c < NCH; ++c) {
            s_fIn[m][c] = (unsigned char)f;
            const int li = s_last[m][c];
            if (li != 255) { any = 1; f = imin(31 - li, 7); }
            else           { f = 7; }                  // chunk len 32 saturates
        }
        s_hasAny[m] = any;
    } else if (tid <= 8) {
        const int m = tid - 5;
        int g = 7;
        for (int c = NCH - 1; c >= 0; --c) {
            s_bIn[m][c] = (unsigned char)g;
            const int fi = s_first[m][c];
            g = (fi != 255) ? imin(fi, 7) : 7;
        }
    }
    __syncthreads();

    // ---- Phase 4: replay chunk fwd (pack sat-distances) then bwd (loss) ----
    float acc = 0.f, cnt = 0.f;
    {
        const int base = tid * CHUNK;
        const int st0  = s_inSt[tid];
        int pm = st0 & 1, tm = (st0 >> 1) & 1;
        int f0 = s_fIn[0][tid], f1 = s_fIn[1][tid], f2 = s_fIn[2][tid], f3 = s_fIn[3][tid];
        unsigned short rec[CHUNK];

#pragma unroll
        for (int p = 0; p < CHUNK; ++p) {
            const int code  = s_code[base + p];
            const int pred  = code & 3;
            const int valid = (code >> 2) & 1;
            const int lab   = (code >> 3) & 3;
            const int m0 = (pred == 2), m1 = (pred == 3);
            const int m2 = valid & (lab == 2), m3 = valid & (lab == 3);

            const int bad   = ((pred == 2) & (pm == 0)) | ((pred == 3) & (pm == 1));
            const int good  = (((lab == 2) & (tm == 1) & (pred == 2)) |
                               ((lab == 3) & (tm == 0) & (pred == 3))) & valid;
            const int mcode = valid ? ((bad << 1) | good) : 0;
            if (valid) {
                pm = (pred == 0 || pred == 2) ? 0 : 1;
                tm = (lab == 0 || lab == 2) ? 0 : 1;
            }
            f0 = m0 ? 0 : imin(f0 + 1, 7);
            f1 = m1 ? 0 : imin(f1 + 1, 7);
            f2 = m2 ? 0 : imin(f2 + 1, 7);
            f3 = m3 ? 0 : imin(f3 + 1, 7);
            rec[p] = (unsigned short)(f0 | (f1 << 3) | (f2 << 6) | (f3 << 9) | (mcode << 12));
        }

        int g0 = s_bIn[0][tid], g1 = s_bIn[1][tid], g2 = s_bIn[2][tid], g3 = s_bIn[3][tid];
        const int ha0 = s_hasAny[0], ha1 = s_hasAny[1], ha2 = s_hasAny[2], ha3 = s_hasAny[3];

#pragma unroll
        for (int p = CHUNK - 1; p >= 0; --p) {
            const int code  = s_code[base + p];
            const int pred  = code & 3;
            const int valid = (code >> 2) & 1;
            const int lab   = (code >> 3) & 3;
            const int m0 = (pred == 2), m1 = (pred == 3);
            const int m2 = valid & (lab == 2), m3 = valid & (lab == 3);

            g0 = m0 ? 0 : imin(g0 + 1, 7);
            g1 = m1 ? 0 : imin(g1 + 1, 7);
            g2 = m2 ? 0 : imin(g2 + 1, 7);
            g3 = m3 ? 0 : imin(g3 + 1, 7);

            const int r  = rec[p];
            const int d0 = imin(r & 7, g0);          // nearest pred==2
            const int d1 = imin((r >> 3) & 7, g1);   // nearest pred==3
            const int d2 = imin((r >> 6) & 7, g2);   // nearest true==2
            const int d3 = imin((r >> 9) & 7, g3);   // nearest true==3

            float mult = s_mode[(r >> 12) & 3];
            if (m0) mult *= ha2 ? s_tab[d2] : 20.0f;             // FP_PEN*2 if no true switch
            if (m2) mult *= ha0 ? ((d0 > TOLD) ? 2.0f : 1.0f) : 3.0f;
            if (m1) mult *= ha3 ? s_tab[d3] : 20.0f;
            if (m3) mult *= ha1 ? ((d1 > TOLD) ? 2.0f : 1.0f) : 3.0f;

            if (valid) { acc += s_ce[base + p] * mult; cnt += 1.0f; }
        }
    }

    // ---- workgroup reduction + global accumulate ----
    s_red[tid]  = acc;
    s_redc[tid] = cnt;
    __syncthreads();
#pragma unroll
    for (int s = 128; s > 0; s >>= 1) {
        if (tid < s) { s_red[tid] += s_red[tid + s]; s_redc[tid] += s_redc[tid + s]; }
        __syncthreads();
    }
    if (tid == 0) {
        atomicAdd(&g_acc[0], s_red[0]);
        atomicAdd(&g_acc[1], s_redc[0]);
    }
}

extern "C" void kernel_launch(void* const* d_in, const int* in_sizes, int n_in,
                              void* d_out, int out_size, void* d_ws, size_t ws_size,
                              hipStream_t stream) {
    const float* logits = (const float*)d_in[0];
    const int*   labels = (const int*)d_in[1];
    float*       out    = (float*)d_out;
    float*       acc    = (float*)d_ws;

    const int B = in_sizes[1] / S_LEN;   // labels are B*S

    hipLaunchKernelGGL(pal_init_acc, dim3(1), dim3(1), 0, stream, acc);
    hipLaunchKernelGGL(pal_row_kernel, dim3(B), dim3(256), 0, stream, logits, labels, acc);
    hipLaunchKernelGGL(pal_finalize, dim3(1), dim3(1), 0, stream, acc, out);
}